// OBBLoss_11854109737210
// MI455X (gfx1250) — compile-verified
//
#include <hip/hip_runtime.h>
#include <stdint.h>

// ---------------------------------------------------------------------------
// OBB IoU loss (1 - IoU of convex quads, Sutherland-Hodgman clip), masked mean.
// CDNA5 path: TDM (tensor_load_to_lds) stages each block's pred/gt tile into
// LDS (TENSORcnt / s_wait_tensorcnt), clip polygons live in bank-conflict-free
// LDS (current vertex carried in VGPRs), wave32 shfl reductions, deterministic
// two-stage final reduce.
// ---------------------------------------------------------------------------

#define TPB 256
#define PAIRS_PER_BLOCK 256
#define TILE_ELEMS (PAIRS_PER_BLOCK * 8)   // 2048 floats per tensor per block

typedef unsigned int u32x4 __attribute__((ext_vector_type(4)));
typedef int          i32x8 __attribute__((ext_vector_type(8)));
typedef int          i32x4 __attribute__((ext_vector_type(4)));

#if defined(__AMDGCN__) && __has_builtin(__builtin_amdgcn_tensor_load_to_lds)
#define USE_TDM 1
#else
#define USE_TDM 0
#endif

__device__ __forceinline__ float quad_area4(const float* x, const float* y) {
  float s = x[0]*y[1] - y[0]*x[1]
          + x[1]*y[2] - y[1]*x[2]
          + x[2]*y[3] - y[2]*x[3]
          + x[3]*y[0] - y[3]*x[0];
  return 0.5f * s;
}

__global__ __launch_bounds__(TPB)
void obb_loss_kernel(const float* __restrict__ pred,
                     const float* __restrict__ gt,
                     const int* __restrict__ img,          // (B,2) [W,H]
                     const unsigned char* __restrict__ vmask,
                     float2* __restrict__ partials,        // (nb) {sum,count}
                     long long total, int N)
{
  __shared__ float  tile[2 * TILE_ELEMS];   // [0..2047]=pred, [2048..4095]=gt
  __shared__ float2 pbufA[8 * TPB];         // polygon ping  [vert][thread]
  __shared__ float2 pbufB[8 * TPB];         // polygon pong
  __shared__ float  wsum[TPB / 32];
  __shared__ float  wcnt[TPB / 32];

  const int tid = threadIdx.x;
  const long long base = (long long)blockIdx.x * PAIRS_PER_BLOCK;
  const long long pair = base + tid;

  // ---- Stage this block's tiles (HBM -> LDS) --------------------------------
  long long remPairs = total - base;
  int remElems = 0;
  if (remPairs > 0)
    remElems = remPairs >= PAIRS_PER_BLOCK ? TILE_ELEMS : (int)(remPairs * 8);

#if USE_TDM
  if (tid < 32 && remElems > 0) {          // wave 0 issues both TDM descriptors
    uint64_t pAddr = (uint64_t)(uintptr_t)(pred + base * 8);
    uint64_t gAddr = (uint64_t)(uintptr_t)(gt   + base * 8);
    uint32_t ldsP  = (uint32_t)(uintptr_t)(&tile[0]);
    uint32_t ldsG  = (uint32_t)(uintptr_t)(&tile[TILE_ELEMS]);
    unsigned d0 = (unsigned)remElems;      // one row of d0 4-byte elements
    // D# group 1: workgroup_mask=0, data_size=2(4B), tensor 1 x d0, tile 1 x d0
    i32x8 g1;
    g1[0] = (int)(2u << 16);                     // data_size = 4 bytes
    g1[1] = (int)((d0 & 0xFFFFu) << 16);         // tensor_dim0[15:0]
    g1[2] = (int)((d0 >> 16) | (1u << 16));      // tensor_dim0[31:16], tensor_dim1=1
    g1[3] = (int)((d0 & 0xFFFFu) << 16);         // tile_dim0 = d0
    g1[4] = 1;                                   // tile_dim1 = 1, tile_dim2 = 0
    g1[5] = (int)d0;                             // tensor_dim0_stride[31:0]
    g1[6] = 0;                                   // stride hi / dim1_stride lo
    g1[7] = 0;
    i32x4 z4 = {0, 0, 0, 0};                     // groups 2/3 unused (<=2D)
    i32x8 z8 = {0, 0, 0, 0, 0, 0, 0, 0};         // extra group (clang-23 form)
    // D# group 0: count=1, lds_addr, global_addr[56:0], type=2
    u32x4 g0p = { 1u, ldsP, (uint32_t)pAddr,
                  (uint32_t)(pAddr >> 32) | (2u << 30) };
    u32x4 g0g = { 1u, ldsG, (uint32_t)gAddr,
                  (uint32_t)(gAddr >> 32) | (2u << 30) };
    __builtin_amdgcn_tensor_load_to_lds(g0p, g1, z4, z4, z8, 0);
    __builtin_amdgcn_tensor_load_to_lds(g0g, g1, z4, z4, z8, 0);
#if __has_builtin(__builtin_amdgcn_s_wait_tensorcnt)
    __builtin_amdgcn_s_wait_tensorcnt(0);
#else
    asm volatile("s_wait_tensorcnt 0" ::: "memory");
#endif
  }
#else
  for (int i = tid; i < remElems; i += TPB) {
    tile[i]              = pred[base * 8 + i];
    tile[TILE_ELEMS + i] = gt[base * 8 + i];
  }
#endif
  __syncthreads();

  // ---- Per-pair loss --------------------------------------------------------
  float loss = 0.f, cval = 0.f;
  if (pair < total) {
    int b = (int)(pair / N);
    float W = (float)img[2 * b], H = (float)img[2 * b + 1];

    const float4* pv = (const float4*)&tile[tid * 8];
    const float4* gvp = (const float4*)&tile[TILE_ELEMS + tid * 8];
    float4 p0 = pv[0], p1 = pv[1];
    float4 q0 = gvp[0], q1 = gvp[1];

    float px[4] = { p0.x * W, p0.z * W, p1.x * W, p1.z * W };
    float py[4] = { p0.y * H, p0.w * H, p1.y * H, p1.w * H };
    float gx[4] = { q0.x * W, q0.z * W, q1.x * W, q1.z * W };
    float gy[4] = { q0.y * H, q0.w * H, q1.y * H, q1.w * H };

    float ag_signed = quad_area4(gx, gy);
    if (ag_signed < 0.f) {                 // g = g[::-1]  (make clip poly CCW)
      float t;
      t = gx[0]; gx[0] = gx[3]; gx[3] = t;  t = gy[0]; gy[0] = gy[3]; gy[3] = t;
      t = gx[1]; gx[1] = gx[2]; gx[2] = t;  t = gy[1]; gy[1] = gy[2]; gy[2] = t;
    }

    float2* pa = pbufA;                    // [v*TPB + tid], conflict-free SoA
    float2* pb = pbufB;
    #pragma unroll
    for (int v = 0; v < 4; ++v) pa[v * TPB + tid] = make_float2(px[v], py[v]);
    int cnt = 4;

    #pragma unroll
    for (int k = 0; k < 4; ++k) {          // clip against 4 gt edges
      float ax = gx[k],            ay = gy[k];
      float bx = gx[(k + 1) & 3],  by = gy[(k + 1) & 3];
      float ex = bx - ax, ey = by - ay;
      int cs = cnt > 1 ? cnt : 1;
      int ocnt = 0;
      int ci = 0;                          // ci tracks i % cs incrementally
      float2 cur = pa[tid];                // poly[0]; carried across iterations
      for (int i = 0; i < 8; ++i) {
        int ni = ci + 1; if (ni == cs) ni = 0;
        float2 nxt = pa[ni * TPB + tid];   // cur_{i+1} == nxt_i (same index)
        float dc = ex * (cur.y - ay) - ey * (cur.x - ax);
        float dn = ex * (nxt.y - ay) - ey * (nxt.x - ax);
        float denom = dc - dn;
        float dd = fabsf(denom) < 1e-12f ? 1e-12f : denom;
        float t = dc / dd;
        float ipx = cur.x + t * (nxt.x - cur.x);
        float ipy = cur.y + t * (nxt.y - cur.y);
        bool active = i < cnt;
        bool ccur = dc >= 0.f;
        bool cnxt = dn >= 0.f;
        if (active && ccur) {
          int o = ocnt < 7 ? ocnt : 7;     // mirror JAX .at[] OOB clamp
          pb[o * TPB + tid] = cur;
          ocnt++;
        }
        if (active && (ccur != cnxt)) {
          int o = ocnt < 7 ? ocnt : 7;
          pb[o * TPB + tid] = make_float2(ipx, ipy);
          ocnt++;
        }
        cur = nxt;
        ci = ni;
      }
      float2* tsw = pa; pa = pb; pb = tsw;
      cnt = ocnt > 8 ? 8 : ocnt;           // MAXV cap
    }

    // shoelace area of the clipped polygon
    float s = 0.f;
    if (cnt > 0) {
      float2 first = pa[tid];
      float2 prev = first;
      for (int i = 1; i < cnt; ++i) {
        float2 v = pa[i * TPB + tid];
        s += prev.x * v.y - prev.y * v.x;
        prev = v;
      }
      s += prev.x * first.y - prev.y * first.x;
    }
    float inter = fabsf(0.5f * s);
    float ap  = fabsf(quad_area4(px, py));
    float ag  = fabsf(ag_signed);
    float uni = ap + ag - inter;
    float iou = uni > 0.f ? inter / uni : 0.f;
    if (vmask[pair]) { loss = 1.f - iou; cval = 1.f; }
  }

  // ---- Deterministic block reduction (wave32 shfl tree + LDS) ---------------
  for (int off = 16; off > 0; off >>= 1) {
    loss += __shfl_down(loss, off, 32);
    cval += __shfl_down(cval, off, 32);
  }
  if ((tid & 31) == 0) { wsum[tid >> 5] = loss; wcnt[tid >> 5] = cval; }
  __syncthreads();
  if (tid == 0) {
    float S = 0.f, C = 0.f;
    #pragma unroll
    for (int w = 0; w < TPB / 32; ++w) { S += wsum[w]; C += wcnt[w]; }
    partials[blockIdx.x] = make_float2(S, C);   // single b64 store
  }
}

__global__ __launch_bounds__(TPB)
void obb_reduce_kernel(const float2* __restrict__ partials, int nb,
                       float* __restrict__ out)
{
  __shared__ float ws[TPB / 32], wc[TPB / 32];
  float s = 0.f, c = 0.f;
  for (int i = threadIdx.x; i < nb; i += TPB) {   // fixed order -> deterministic
    float2 p = partials[i];
    s += p.x;
    c += p.y;
  }
  for (int off = 16; off > 0; off >>= 1) {
    s += __shfl_down(s, off, 32);
    c += __shfl_down(c, off, 32);
  }
  if ((threadIdx.x & 31) == 0) { ws[threadIdx.x >> 5] = s; wc[threadIdx.x >> 5] = c; }
  __syncthreads();
  if (threadIdx.x == 0) {
    float S = 0.f, C = 0.f;
    #pragma unroll
    for (int w = 0; w < TPB / 32; ++w) { S += ws[w]; C += wc[w]; }
    float mean = S / fmaxf(C, 1.0f);
    out[0] = (C > 0.f) ? mean : 0.f;
  }
}

extern "C" void kernel_launch(void* const* d_in, const int* in_sizes, int n_in,
                              void* d_out, int out_size, void* d_ws, size_t ws_size,
                              hipStream_t stream) {
  const float* pred        = (const float*)d_in[0];   // (B,N,8) f32
  const float* gt          = (const float*)d_in[1];   // (B,N,8) f32
  const int*   img         = (const int*)d_in[2];     // (B,2) i32
  const unsigned char* vm  = (const unsigned char*)d_in[3]; // (B,N) bool

  int B = in_sizes[2] / 2;
  long long total = (long long)in_sizes[0] / 8;       // B*N pairs
  int N = (int)(total / (B > 0 ? B : 1));
  int nb = (int)((total + PAIRS_PER_BLOCK - 1) / PAIRS_PER_BLOCK);

  float2* partials = (float2*)d_ws;                   // nb float2 of scratch

  obb_loss_kernel<<<nb, TPB, 0, stream>>>(pred, gt, img, vm, partials, total, N);
  obb_reduce_kernel<<<1, TPB, 0, stream>>>(partials, nb, (float*)d_out);
}